// FireflyAlgorithm_50964081934827
// MI455X (gfx1250) — compile-verified
//
#include <hip/hip_runtime.h>
#include <math.h>

typedef __attribute__((ext_vector_type(2))) float v2f;
typedef __attribute__((ext_vector_type(8))) float v8f;

#define NF   128   // fireflies
#define ND   2048  // dims
#define NB   8     // batch
#define TPB  256   // threads per block (8 wave32)
#define PER_T (ND / TPB)  // 8 floats per thread

// ---------------------------------------------------------------------------
// Half-wave swap: every lane gets the value held by lane (L ^ 16).
// v_permlanex16_b32 with identity selects crosses the two 16-lane rows of a
// wave32 in a single VALU op (replaces the ds_bpermute the generic shuffle
// lowers to -- keeps the per-step reduction off the LDS pipe).
// ---------------------------------------------------------------------------
__device__ __forceinline__ float xor16_swap(float v) {
  int i = __builtin_bit_cast(int, v);
  int o = __builtin_amdgcn_permlanex16(i, i, 0x76543210, 0xfedcba98, false, false);
  return __builtin_bit_cast(float, o);
}

// ---------------------------------------------------------------------------
// Wave32 all-lane sum using V_WMMA_F32_16X16X4_F32.
// A layout (16x4 f32): lanes 0-15 hold {K=0,K=1}, lanes 16-31 hold {K=2,K=3}.
// Put each lane's partial in its first slot, zero the other; B = all ones
// (every element 1 regardless of B's striping). Then D[m][n] = p_m + p_{m+16}.
// C/D layout: lane<16 has rows M=vgpr (0..7), lane>=16 rows M=vgpr+8.
// Sum of a lane's 8 accumulators = half-sum; xor16_swap adds the other half.
// EXEC must be all ones (guaranteed: all caller branches are block-uniform).
// ---------------------------------------------------------------------------
__device__ __forceinline__ float wave_sum32(float p) {
  v2f a; a[0] = p;    a[1] = 0.0f;
  v2f b; b[0] = 1.0f; b[1] = 1.0f;
  v8f c = {};
  c = __builtin_amdgcn_wmma_f32_16x16x4_f32(false, a, false, b,
                                            (short)0, c, false, false);
  float s = ((c[0] + c[1]) + (c[2] + c[3])) + ((c[4] + c[5]) + (c[6] + c[7]));
  s += xor16_swap(s);
  return s;
}

// Block-wide (256-thread) sum; result valid on thread 0.
__device__ __forceinline__ float block_sum(float p, float* sm8) {
  float s = wave_sum32(p);
  int t = threadIdx.x;
  if ((t & 31) == 0) sm8[t >> 5] = s;
  __syncthreads();
  float tot = 0.0f;
#pragma unroll
  for (int w = 0; w < 8; ++w) tot += sm8[w];
  return tot;
}

// ---------------------------------------------------------------------------
// Kernel 1: copy positions into workspace and compute squared-norm snapshot
// (the stale "intensities" the loop conditions read). One block per row.
// ---------------------------------------------------------------------------
__global__ void __launch_bounds__(TPB) ff_init(const float* __restrict__ pos_in,
                                               float* __restrict__ pos,
                                               float* __restrict__ intens) {
  __shared__ float sm8[8];
  const int r = blockIdx.x, t = threadIdx.x;
  const float4* src = reinterpret_cast<const float4*>(pos_in + (size_t)r * ND) + t * 2;
  float4*       dst = reinterpret_cast<float4*>(pos + (size_t)r * ND) + t * 2;
  float4 v0 = src[0], v1 = src[1];
  dst[0] = v0; dst[1] = v1;
  float p = v0.x*v0.x + v0.y*v0.y + v0.z*v0.z + v0.w*v0.w
          + v1.x*v1.x + v1.y*v1.y + v1.z*v1.z + v1.w*v1.w;
  float tot = block_sum(p, sm8);
  if (t == 0) intens[r] = tot;   // squared norm (monotone-equivalent for compares)
}

// ---------------------------------------------------------------------------
// Kernel 2: the sequential firefly scan. Single block of 8 wave32s.
// Thread t owns dims [t*8, t*8+8). The accept test depends only on the stale
// snapshot, so it is block-uniform: EXEC stays all-ones (WMMA-legal), skipped
// pairs cost nothing, and the accepted-j sequence is known ahead of time --
// which lets us software-pipeline the next step's pos_j/noise loads so their
// L2/HBM latency overlaps the current step's barrier + reduce + exp.
// One barrier per accepted step via parity-double-buffered wave-sum slots.
// ---------------------------------------------------------------------------
__global__ void __launch_bounds__(TPB, 1) ff_main(float* __restrict__ pos,
                                                  const float* __restrict__ noise,
                                                  const float* __restrict__ intens_g) {
  const int t = threadIdx.x;
  const int wave = t >> 5;
  __shared__ float intens_s[NF];
  __shared__ alignas(16) float wsum[16];   // 8 waves x 2 parities
  if (t < NF) intens_s[t] = intens_g[t];
  __syncthreads();

  int ph = 0;
  for (int i = 0; i < NF; ++i) {
    float* rowi = pos + (size_t)i * ND + t * PER_T;
    float pi[PER_T];
    {
      float4 a = *reinterpret_cast<const float4*>(rowi);
      float4 b = *reinterpret_cast<const float4*>(rowi + 4);
      pi[0]=a.x; pi[1]=a.y; pi[2]=a.z; pi[3]=a.w;
      pi[4]=b.x; pi[5]=b.y; pi[6]=b.z; pi[7]=b.w;
    }
    const float inti = intens_s[i];
    const float* noise_i = noise + (size_t)i * NF * ND + t * PER_T;

    // First accepted j (uniform scan of the stale snapshot).
    int j = 0;
    while (j < NF && !((j != i) && (intens_s[j] < inti))) ++j;

    float4 pj0, pj1, n0, n1;   // data for the step being computed
    if (j < NF) {
      const float* rowj = pos + (size_t)j * ND + t * PER_T;
      const float* nz   = noise_i + (size_t)j * ND;
      pj0 = *reinterpret_cast<const float4*>(rowj);
      pj1 = *reinterpret_cast<const float4*>(rowj + 4);
      n0  = *reinterpret_cast<const float4*>(nz);
      n1  = *reinterpret_cast<const float4*>(nz + 4);
    }

    while (j < NF) {
      // --- issue next accepted step's loads before this step's barrier ---
      int jn = j + 1;
      while (jn < NF && !((jn != i) && (intens_s[jn] < inti))) ++jn;
      float4 qj0, qj1, m0, m1;
      if (jn < NF) {
        const float* rowjn = pos + (size_t)jn * ND + t * PER_T;
        const float* nzn   = noise_i + (size_t)jn * ND;
        qj0 = *reinterpret_cast<const float4*>(rowjn);
        qj1 = *reinterpret_cast<const float4*>(rowjn + 4);
        m0  = *reinterpret_cast<const float4*>(nzn);
        m1  = *reinterpret_cast<const float4*>(nzn + 4);
        if (jn + 1 < NF)
          __builtin_prefetch(nzn + ND, 0, 1);   // global_prefetch_b8: next noise row
      }

      // --- current step: d2 = ||pi - pj||^2 via WMMA wave reduce + LDS ---
      float pj[PER_T] = {pj0.x,pj0.y,pj0.z,pj0.w, pj1.x,pj1.y,pj1.z,pj1.w};
      float nn[PER_T] = {n0.x, n0.y, n0.z, n0.w,  n1.x, n1.y, n1.z, n1.w };

      float part = 0.0f;
#pragma unroll
      for (int k = 0; k < PER_T; ++k) {
        float d = pi[k] - pj[k];
        part = fmaf(d, d, part);
      }
      float s = wave_sum32(part);                  // v_wmma_f32_16x16x4_f32
      if ((t & 31) == 0) wsum[ph * 8 + wave] = s;
      __syncthreads();
      float4 w0 = reinterpret_cast<float4*>(wsum)[ph * 2];      // ds_load_b128
      float4 w1 = reinterpret_cast<float4*>(wsum)[ph * 2 + 1];
      ph ^= 1;
      float d2 = ((w0.x + w0.y) + (w0.z + w0.w)) + ((w1.x + w1.y) + (w1.z + w1.w));

      float beta = __expf(-d2);                    // GAMMA = BETA0 = 1
#pragma unroll
      for (int k = 0; k < PER_T; ++k)
        pi[k] = fmaf(beta, pj[k] - pi[k], fmaf(0.2f, nn[k], pi[k]));

      // rotate pipeline
      pj0 = qj0; pj1 = qj1; n0 = m0; n1 = m1;
      j = jn;
    }

    *reinterpret_cast<float4*>(rowi)     = make_float4(pi[0], pi[1], pi[2], pi[3]);
    *reinterpret_cast<float4*>(rowi + 4) = make_float4(pi[4], pi[5], pi[6], pi[7]);
    __syncthreads();
  }
}

// ---------------------------------------------------------------------------
// Kernel 3: squared norms of the updated rows.
// ---------------------------------------------------------------------------
__global__ void __launch_bounds__(TPB) ff_norms(const float* __restrict__ pos,
                                                float* __restrict__ intens2) {
  __shared__ float sm8[8];
  const int r = blockIdx.x, t = threadIdx.x;
  const float4* src = reinterpret_cast<const float4*>(pos + (size_t)r * ND) + t * 2;
  float4 v0 = src[0], v1 = src[1];
  float p = v0.x*v0.x + v0.y*v0.y + v0.z*v0.z + v0.w*v0.w
          + v1.x*v1.x + v1.y*v1.y + v1.z*v1.z + v1.w*v1.w;
  float tot = block_sum(p, sm8);
  if (t == 0) intens2[r] = tot;
}

// ---------------------------------------------------------------------------
// Kernel 4: uniform argmin over 128 squared norms (first-min, matching
// jnp.argmin), conditional select vs best_position, broadcast to [8, 2048].
// ---------------------------------------------------------------------------
__global__ void __launch_bounds__(TPB) ff_out(const float* __restrict__ pos,
                                              const float* __restrict__ intens2,
                                              const float* __restrict__ best_position,
                                              const float* __restrict__ best_intensity,
                                              float* __restrict__ out) {
  const int d = blockIdx.x * TPB + threadIdx.x;   // 0..2047, uniform loop below
  float mn = intens2[0];
  int   mi = 0;
  for (int r = 1; r < NF; ++r) {
    float v = intens2[r];
    if (v < mn) { mn = v; mi = r; }
  }
  const bool better = sqrtf(mn) < best_intensity[0];
  const float val = better ? pos[(size_t)mi * ND + d] : best_position[d];
#pragma unroll
  for (int b = 0; b < NB; ++b) out[(size_t)b * ND + d] = val;
}

// ---------------------------------------------------------------------------
// Host entry. Inputs (setup_inputs order): x, positions, noise, best_position,
// best_intensity. Output: [8, 2048] f32. Workspace: 1 MB positions copy +
// two 128-float norm arrays.
// ---------------------------------------------------------------------------
extern "C" void kernel_launch(void* const* d_in, const int* in_sizes, int n_in,
                              void* d_out, int out_size, void* d_ws, size_t ws_size,
                              hipStream_t stream) {
  (void)in_sizes; (void)n_in; (void)out_size; (void)ws_size;
  const float* positions      = (const float*)d_in[1];
  const float* noise          = (const float*)d_in[2];
  const float* best_position  = (const float*)d_in[3];
  const float* best_intensity = (const float*)d_in[4];
  float* out = (float*)d_out;

  float* ws_pos     = (float*)d_ws;          // NF*ND floats
  float* ws_intens  = ws_pos + NF * ND;      // NF floats (stale snapshot)
  float* ws_intens2 = ws_intens + NF;        // NF floats (final)

  ff_init <<<NF, TPB, 0, stream>>>(positions, ws_pos, ws_intens);
  ff_main <<<1,  TPB, 0, stream>>>(ws_pos, noise, ws_intens);
  ff_norms<<<NF, TPB, 0, stream>>>(ws_pos, ws_intens2);
  ff_out  <<<ND / TPB, TPB, 0, stream>>>(ws_pos, ws_intens2,
                                         best_position, best_intensity, out);
}